// BiInteraction_38577396253196
// MI455X (gfx1250) — compile-verified
//
#include <hip/hip_runtime.h>

// ---------------------------------------------------------------------------
// BiInteraction on MI455X (gfx1250).
//   B=8192 rows, N=34 fields, D=16.  out[b, p*16+d] = (x[b,i,:]·W)[d] * x[b,j,d]
// Roofline: 0.22 GFLOP vs 311 MB stores -> store-bandwidth bound (~14us floor
// at 23.3 TB/s). Use V_WMMA_F32_16X16X4_F32 for the 16x16 projection (keeps
// proj on-chip in LDS), then stream coalesced NT b128 stores for the output.
// ---------------------------------------------------------------------------

typedef __attribute__((ext_vector_type(2))) float v2f;
typedef __attribute__((ext_vector_type(4))) float v4f;
typedef __attribute__((ext_vector_type(8))) float v8f;

#define NF        34          // fields
#define DD        16          // embedding dim
#define NPAIR     594         // sum_{i=0}^{32} (34-i)
#define ROWF      (NPAIR*DD)  // 9504 floats per output row
#define ROWV4     (ROWF/4)    // 2376 float4 per output row
#define TROWS     16          // batch rows per wave (one WMMA tile)
#define WPB       4           // waves per block
#define NPROJ     33          // i-side only uses fields 0..32

// Compile-time (i,j) pair table, packed i | (j<<8) into a u16.
struct PairTab { unsigned short v[NPAIR]; };
static constexpr PairTab makeTab() {
    PairTab t{};
    int p = 0;
    for (int i = 0; i < NF - 1; ++i)
        for (int j = i; j < NF; ++j)
            t.v[p++] = (unsigned short)(i | (j << 8));
    return t;
}
__device__ __constant__ PairTab kTab = makeTab();

__global__ __launch_bounds__(WPB * 32)
void BiInteraction_38577396253196_kernel(const float* __restrict__ x,
                                         const float* __restrict__ W,
                                         float* __restrict__ out) {
    extern __shared__ float lds[];              // WPB * NPROJ * 256 floats
    const int lane = threadIdx.x & 31;
    const int wave = threadIdx.x >> 5;
    const int hi   = lane >> 4;                 // half-wave: 0 or 1
    const int lo   = lane & 15;
    const int b0   = (blockIdx.x * WPB + wave) * TROWS;   // batch tile base
    float* pl = lds + wave * (NPROJ * 256);     // this wave's private proj slice

    // ---- Stage W (16x16) in B-operand layout, 4 K-chunks of 4 -------------
    // B 4x16 f32: VGPR0 = {K=k0 | K=k0+2} across half-waves, VGPR1 = {k0+1 | k0+3}.
    v2f wb[4];
#pragma unroll
    for (int c = 0; c < 4; ++c) {
        const int k = 4 * c + 2 * hi;
        wb[c].x = W[k * DD + lo];
        wb[c].y = W[(k + 1) * DD + lo];
    }

    // ---- Phase 1: proj[n] = X_tile(16x16) * W via 4x V_WMMA_F32_16X16X4_F32
    for (int n = 0; n < NPROJ; ++n) {
        const float* xa = x + ((size_t)(b0 + lo) * NF + n) * DD;
        v8f acc = {};
#pragma unroll
        for (int c = 0; c < 4; ++c) {
            // A 16x4 f32: lane L<16 holds (k0,k0+1) of row L; L>=16 holds (k0+2,k0+3).
            v2f a = *(const v2f*)(xa + 4 * c + 2 * hi);
            acc = __builtin_amdgcn_wmma_f32_16x16x4_f32(
                /*neg_a=*/false, a, /*neg_b=*/false, wb[c],
                /*c_mod=*/(short)0, acc, /*reuse_a=*/false, /*reuse_b=*/false);
        }
        // C/D layout: VGPR v, lanes 0-15 -> row v, lanes 16-31 -> row v+8; col = lo.
#pragma unroll
        for (int v = 0; v < 8; ++v)
            pl[n * 256 + (v + 8 * hi) * TROWS + lo] = acc[v];
    }
    // No barrier: each wave reads only its own LDS slice; DS ops are in-order
    // per wave and the compiler inserts the dscnt waits.

    // ---- Phase 2: stream 594 pairwise products per row, NT b128 stores ----
    for (int r = 0; r < TROWS; ++r) {
        const float* xrow = x   + (size_t)(b0 + r) * (NF * DD);
        float*       orow = out + (size_t)(b0 + r) * ROWF;
        const float* plr  = pl + r * DD;        // proj[n][r][:] at plr + n*256
        for (int c = 0; c < (ROWV4 + WPB * 0 + 31) / 32; ++c) {   // 75 iters
            const int e = c * 32 + lane;        // float4 index within row
            if (e < ROWV4) {
                const int pp = e >> 2;          // pair index
                const int d0 = (e & 3) * 4;     // dim sub-offset
                const unsigned pk = kTab.v[pp];
                const int i = pk & 0xFF;
                const int j = pk >> 8;
                v4f pv = *(const v4f*)(plr + i * 256 + d0);   // LDS, 16B aligned
                v4f xv = *(const v4f*)(xrow + j * DD + d0);   // global, L2-hot
                v4f o  = pv * xv;
                // Output never re-read and > L2 capacity: non-temporal store.
                __builtin_nontemporal_store(o, (v4f*)(orow + pp * DD + d0));
            }
        }
    }
}

extern "C" void kernel_launch(void* const* d_in, const int* in_sizes, int n_in,
                              void* d_out, int out_size, void* d_ws, size_t ws_size,
                              hipStream_t stream) {
    const float* x = (const float*)d_in[0];   // [8192, 34, 16] f32
    const float* W = (const float*)d_in[1];   // [16, 16] f32
    float* out = (float*)d_out;               // [8192, 9504] f32

    const int B = 8192;
    const int blocks = B / (TROWS * WPB);     // 128 blocks of 4 waves
    const size_t ldsBytes = (size_t)WPB * NPROJ * 256 * sizeof(float); // 132 KB < 320 KB/WGP
    hipLaunchKernelGGL(BiInteraction_38577396253196_kernel,
                       dim3(blocks), dim3(WPB * 32), ldsBytes, stream,
                       x, W, out);
}